// OctreeMorton_33217277067306
// MI455X (gfx1250) — compile-verified
//
#include <hip/hip_runtime.h>
#include <stdint.h>

typedef __attribute__((ext_vector_type(2))) float v2f;
typedef __attribute__((ext_vector_type(8))) float v8f;

// descriptor / payload vector types matching builtin signatures
typedef unsigned int u32x4 __attribute__((__vector_size__(16)));
typedef int i32x4 __attribute__((__vector_size__(16)));
typedef int i32x8 __attribute__((__vector_size__(32)));
typedef int v4i  __attribute__((__vector_size__(16)));

typedef __attribute__((address_space(1))) v4i gv4i;   // global-typed 16B vector
typedef __attribute__((address_space(3))) v4i lv4i;   // LDS-typed 16B vector
typedef __attribute__((address_space(3))) void as3_void;

#define SENT_LL  (1LL << 62)
#define SENT_ULL (1ULL << 62)

#if __has_builtin(__builtin_amdgcn_global_load_async_to_lds_b128) && __has_builtin(__builtin_amdgcn_s_wait_asynccnt)
#define HAVE_ASYNC 1
#else
#define HAVE_ASYNC 0
#endif

#if __has_builtin(__builtin_amdgcn_tensor_load_to_lds) && __has_builtin(__builtin_amdgcn_s_wait_tensorcnt)
#define HAVE_TDM 1
#else
#define HAVE_TDM 0
#endif

// ---------------- helpers ----------------

__device__ __forceinline__ unsigned long long split_by3(unsigned long long n) {
    n &= 0x1FFFFFULL;
    n = (n | (n << 32)) & 0x001F00000000FFFFULL;
    n = (n | (n << 16)) & 0x001F0000FF0000FFULL;
    n = (n | (n << 8))  & 0x100F00F00F00F00FULL;
    n = (n | (n << 4))  & 0x10C30C30C30C30C3ULL;
    n = (n | (n << 2))  & 0x1249249249249249ULL;
    return n;
}

__device__ __forceinline__ unsigned fenc(float f) {
    unsigned b = __float_as_uint(f);
    return (b & 0x80000000u) ? ~b : (b | 0x80000000u);
}
__device__ __forceinline__ float fdec(unsigned u) {
    unsigned b = (u & 0x80000000u) ? (u ^ 0x80000000u) : ~u;
    return __uint_as_float(b);
}

// ---------------- min reduction ----------------

__global__ void k_init_min(unsigned* minb) {
    if (threadIdx.x < 3) minb[threadIdx.x] = 0xFFFFFFFFu;
}

__global__ void k_min(const float* __restrict__ cloud, int n, unsigned* minb) {
    __shared__ unsigned sm[3][256];
    int tid = threadIdx.x;
    int i = blockIdx.x * 256 + tid;
    unsigned e0 = 0xFFFFFFFFu, e1 = 0xFFFFFFFFu, e2 = 0xFFFFFFFFu;
    if (i < n) {
        size_t b = (size_t)i * 4;
        e0 = fenc(cloud[b + 0]);
        e1 = fenc(cloud[b + 1]);
        e2 = fenc(cloud[b + 2]);
    }
    sm[0][tid] = e0; sm[1][tid] = e1; sm[2][tid] = e2;
    __syncthreads();
    for (int s = 128; s > 0; s >>= 1) {
        if (tid < s) {
            for (int c = 0; c < 3; ++c) {
                unsigned a = sm[c][tid], bb = sm[c][tid + s];
                sm[c][tid] = a < bb ? a : bb;
            }
        }
        __syncthreads();
    }
    if (tid == 0) {
        atomicMin(&minb[0], sm[0][0]);
        atomicMin(&minb[1], sm[1][0]);
        atomicMin(&minb[2], sm[2][0]);
    }
}

// ---------------- morton + feats ----------------

__global__ void k_morton(const float* __restrict__ cloud, const float* __restrict__ leaf,
                         const unsigned* __restrict__ minb, int n,
                         unsigned long long* __restrict__ codes,
                         float* __restrict__ feats, unsigned* __restrict__ idx) {
    int i = blockIdx.x * 256 + threadIdx.x;
    if (i >= n) return;
    size_t b = (size_t)i * 4;
    float x = cloud[b + 0], y = cloud[b + 1], z = cloud[b + 2], w = cloud[b + 3];
    float lx = leaf[0], ly = leaf[1], lz = leaf[2];
    float ox = fdec(minb[0]) - 0.5f * lx;
    float oy = fdec(minb[1]) - 0.5f * ly;
    float oz = fdec(minb[2]) - 0.5f * lz;
    long long ix = (long long)floorf((x - ox) / lx);
    long long iy = (long long)floorf((y - oy) / ly);
    long long iz = (long long)floorf((z - oz) / lz);
    unsigned long long code = split_by3((unsigned long long)ix)
                            | (split_by3((unsigned long long)iy) << 1)
                            | (split_by3((unsigned long long)iz) << 2);
    codes[i] = code;
    idx[i] = (unsigned)i;
    size_t f = (size_t)i * 8;
    feats[f + 0] = 1.0f;
    feats[f + 1] = x;  feats[f + 2] = y;  feats[f + 3] = z;
    feats[f + 4] = x * x; feats[f + 5] = y * y; feats[f + 6] = z * z;
    feats[f + 7] = w;
}

// ---------------- radix sort (8 bits x 8 passes) ----------------

__global__ void k_hist(const unsigned long long* __restrict__ kin, int n, int shift,
                       unsigned* __restrict__ counts, int numB) {
    __shared__ unsigned h[256];
    int tid = threadIdx.x;
    h[tid] = 0;
    __syncthreads();
    int i = blockIdx.x * 256 + tid;
    if (i < n) {
        __builtin_prefetch(&kin[i], 0, 1);   // global_prefetch_b8
        unsigned bin = (unsigned)((kin[i] >> shift) & 0xFFULL);
        atomicAdd(&h[bin], 1u);
    }
    __syncthreads();
    counts[(size_t)tid * numB + blockIdx.x] = h[tid];
}

// per-block exclusive scan -> excl partial + block sums (hierarchical scan stage 1)
__global__ void k_bexscan(const unsigned* __restrict__ in, unsigned* __restrict__ outExcl,
                          unsigned* __restrict__ bsums, int m) {
    __shared__ unsigned t[256];
    int tid = threadIdx.x;
    int i = blockIdx.x * 256 + tid;
    unsigned v = (i < m) ? in[i] : 0u;
    t[tid] = v;
    __syncthreads();
    for (int off = 1; off < 256; off <<= 1) {
        unsigned u = (tid >= off) ? t[tid - off] : 0u;
        __syncthreads();
        t[tid] += u;
        __syncthreads();
    }
    if (i < m) outExcl[i] = t[tid] - v;
    if (tid == 255) bsums[blockIdx.x] = t[255];
}

__global__ void k_addscan(const unsigned* __restrict__ excl, const unsigned* __restrict__ bsums,
                          unsigned* __restrict__ out, int m) {
    int i = blockIdx.x * 256 + threadIdx.x;
    if (i < m) out[i] = excl[i] + bsums[blockIdx.x];
}

// single-block chunked exclusive scan, in place (used for block sums only)
__global__ void __launch_bounds__(1024) k_scan_single(unsigned* __restrict__ data, int m) {
    __shared__ unsigned t[1024];
    __shared__ unsigned carry_s;
    int tid = threadIdx.x;
    if (tid == 0) carry_s = 0;
    __syncthreads();
    for (int base = 0; base < m; base += 1024) {
        unsigned v = (base + tid < m) ? data[base + tid] : 0u;
        t[tid] = v;
        __syncthreads();
        for (int off = 1; off < 1024; off <<= 1) {
            unsigned u = (tid >= off) ? t[tid - off] : 0u;
            __syncthreads();
            t[tid] += u;
            __syncthreads();
        }
        unsigned incl = t[tid];
        unsigned carry = carry_s;
        if (base + tid < m) data[base + tid] = (incl - v) + carry;
        __syncthreads();
        if (tid == 1023) carry_s = carry + incl;
        __syncthreads();
    }
}

__global__ void k_scatter(const unsigned long long* __restrict__ kin,
                          const unsigned* __restrict__ iin, int n, int shift,
                          const unsigned* __restrict__ counts, int numB,
                          unsigned long long* __restrict__ kout, unsigned* __restrict__ iout) {
    __shared__ unsigned long long sk[256];
    __shared__ unsigned sb[256];
    int tid = threadIdx.x;
    int i = blockIdx.x * 256 + tid;
    unsigned long long key = 0;
    unsigned bin = 0xFFFFFFFFu;
    if (i < n) {
        key = kin[i];
        bin = (unsigned)((key >> shift) & 0xFFULL);
    }
    sk[tid] = key; sb[tid] = bin;
    __syncthreads();
    if (i < n) {
        unsigned rank = 0;
        for (int j = 0; j < tid; ++j) rank += (sb[j] == bin) ? 1u : 0u;
        unsigned dest = counts[(size_t)bin * numB + blockIdx.x] + rank;
        kout[dest] = key;
        iout[dest] = iin[i];
    }
}

// ---------------- segmented aggregation machinery ----------------

__global__ void k_flags(const unsigned long long* __restrict__ codes, unsigned* __restrict__ flags, int n) {
    int i = blockIdx.x * 256 + threadIdx.x;
    if (i >= n) return;
    flags[i] = (i == 0 || codes[i] != codes[i - 1]) ? 1u : 0u;
}

__global__ void k_blockscan(const unsigned* __restrict__ flags, unsigned* __restrict__ seg,
                            unsigned* __restrict__ bsums, int n) {
    __shared__ unsigned t[256];
    int tid = threadIdx.x;
    int i = blockIdx.x * 256 + tid;
    unsigned v = (i < n) ? flags[i] : 0u;
    t[tid] = v;
    __syncthreads();
    for (int off = 1; off < 256; off <<= 1) {
        unsigned u = (tid >= off) ? t[tid - off] : 0u;
        __syncthreads();
        t[tid] += u;
        __syncthreads();
    }
    if (i < n) seg[i] = t[tid];            // inclusive within block
    if (tid == 255) bsums[blockIdx.x] = t[255];
}

__global__ void k_addoff(unsigned* __restrict__ seg, const unsigned* __restrict__ bsums,
                         const unsigned* __restrict__ flags, unsigned* __restrict__ head, int n) {
    int i = blockIdx.x * 256 + threadIdx.x;
    if (i >= n) return;
    unsigned s = seg[i] + bsums[blockIdx.x] - 1u;   // cumsum(is_new)-1
    seg[i] = s;
    if (flags[i]) head[s] = (unsigned)i;
}

// Per-wave 16-point tile: D(16x16) = S(16x16 one-hot) x F(16x16 feats) via
// four chained V_WMMA_F32_16X16X4_F32. Staging uses TDM (contiguous levels),
// async-to-LDS B128 (gather level), or scalar fallback.
__global__ void k_tile_wmma(const float* __restrict__ vals, const unsigned* __restrict__ gidx,
                            const unsigned* __restrict__ seg, int n,
                            float* __restrict__ tilebuf, unsigned* __restrict__ tilefirst,
                            int numTiles) {
    __shared__ __attribute__((aligned(16))) float lv[8][16][16];  // [wave][k][n]
    __shared__ unsigned ls[8][16];
    int wave = threadIdx.x >> 5;
    int lane = threadIdx.x & 31;
    int tileId = blockIdx.x * 8 + wave;
    if (tileId >= numTiles) return;      // uniform per wave
    int base = tileId * 16;
    unsigned segbase = seg[base];

    // local segment ids (lanes 0-15)
    if (lane < 16) {
        int i = base + lane;
        ls[wave][lane] = (i < n) ? (seg[i] - segbase) : 15u;
    }

    // pre-zero pad columns 8..15 (and whole rows past n) — 4 floats per lane
    {
        int k = lane >> 1, h = lane & 1;
#pragma unroll
        for (int j = 0; j < 4; ++j) lv[wave][k][8 + 4 * h + j] = 0.0f;
        if (base + k >= n) {
#pragma unroll
            for (int j = 0; j < 4; ++j) lv[wave][k][4 * h + j] = 0.0f;
        }
    }

    bool tdm_used = false;
#if HAVE_TDM
    if (gidx == nullptr && base + 16 <= n) {
        // 2D tensor tile: 16 rows x 8 f32, row stride 8; LDS pad 8 DW after
        // every 8 DW -> rows land on the 16-float LDS row stride.
        unsigned ldsoff = (unsigned)(size_t)(as3_void*)&lv[wave][0][0];
        ldsoff = (unsigned)__builtin_amdgcn_readfirstlane((int)ldsoff);
        unsigned long long ga = (unsigned long long)(size_t)vals + (unsigned long long)base * 32ull;
        unsigned galo = (unsigned)__builtin_amdgcn_readfirstlane((int)(unsigned)ga);
        unsigned gahi = (unsigned)__builtin_amdgcn_readfirstlane((int)(unsigned)(ga >> 32));
        u32x4 g0 = { 1u,                                  // count=1, user mode
                     ldsoff,                              // lds_addr
                     galo,                                // global_addr[31:0]
                     (gahi & 0x01FFFFFFu) | 0x80000000u };// global_addr[56:32] | type=2
        i32x8 g1 = { (2 << 16) | (1 << 20) | (2 << 22) | (7 << 25), // data_size=4B, pad_en, int=8DW, amt=8DW
                     (8 << 16),                           // tensor_dim0 = 8 (low bits)
                     (16 << 16),                          // tensor_dim1 = 16 (low bits)
                     (8 << 16),                           // tile_dim0 = 8
                     16,                                  // tile_dim1 = 16
                     8,                                   // tensor_dim0_stride = 8
                     0, 0 };
        i32x4 gz4 = { 0, 0, 0, 0 };
        i32x8 gz8 = { 0, 0, 0, 0, 0, 0, 0, 0 };
        __builtin_amdgcn_tensor_load_to_lds(g0, g1, gz4, gz4, gz8, 0);
        tdm_used = true;
    }
#endif

    if (!tdm_used) {
#if HAVE_ASYNC
        int k = lane >> 1, h = lane & 1;   // lane -> (row, half-row of 4 f32)
        int i = base + k;
        if (i < n) {
            unsigned p = gidx ? gidx[i] : (unsigned)i;
            __builtin_amdgcn_global_load_async_to_lds_b128(
                (gv4i*)const_cast<float*>(vals + (size_t)p * 8 + (size_t)h * 4),
                (lv4i*)&lv[wave][k][h * 4], 0, 0);
        }
#else
        if (lane < 16) {
            int i = base + lane;
            if (i < n) {
                unsigned p = gidx ? gidx[i] : (unsigned)i;
                size_t fo = (size_t)p * 8;
#pragma unroll
                for (int f = 0; f < 8; ++f) lv[wave][lane][f] = vals[fo + f];
            }
        }
#endif
    }

#if HAVE_TDM
    if (tdm_used) __builtin_amdgcn_s_wait_tensorcnt(0);
#endif
#if HAVE_ASYNC
    __builtin_amdgcn_s_wait_asynccnt(0);
#endif
    __syncthreads();

    int half = lane >> 4;     // 0: lanes 0-15, 1: lanes 16-31
    int rc   = lane & 15;     // A row / B col
    v8f acc = {0.f, 0.f, 0.f, 0.f, 0.f, 0.f, 0.f, 0.f};
#pragma unroll
    for (int c = 0; c < 4; ++c) {
        int klo = 4 * c + half * 2;      // VGPR0: K=klo, VGPR1: K=klo+1
        v2f a, b;
        a.x = (ls[wave][klo]     == (unsigned)rc) ? 1.0f : 0.0f;
        a.y = (ls[wave][klo + 1] == (unsigned)rc) ? 1.0f : 0.0f;
        b.x = lv[wave][klo][rc];
        b.y = lv[wave][klo + 1][rc];
        acc = __builtin_amdgcn_wmma_f32_16x16x4_f32(
            /*neg_a=*/false, a, /*neg_b=*/false, b,
            /*c_mod=*/(short)0, acc, /*reuse_a=*/false, /*reuse_b=*/false);
    }

    // D layout: VGPR v -> lanes 0-15: M=v, N=lane; lanes 16-31: M=v+8, N=lane-16
    if (rc < 8) {
        int row0 = half * 8;
        size_t tb = (size_t)tileId * 128;
#pragma unroll
        for (int v = 0; v < 8; ++v)
            tilebuf[tb + (size_t)(row0 + v) * 8 + rc] = acc[v];
    }
    if (lane == 0) tilefirst[tileId] = segbase;
}

__global__ void k_combine(const unsigned long long* __restrict__ codes,
                          const unsigned* __restrict__ seg, const unsigned* __restrict__ head,
                          const float* __restrict__ tilebuf, const unsigned* __restrict__ tilefirst,
                          int n, long long* __restrict__ out_codes, float* __restrict__ out_agg,
                          int* __restrict__ out_cnt) {
    int g = blockIdx.x * 256 + threadIdx.x;
    if (g >= n) return;
    unsigned total = seg[n - 1] + 1u;
    if (g == 0) *out_cnt = (int)total - ((codes[n - 1] == SENT_ULL) ? 1 : 0);
    if ((unsigned)g < total) {
        unsigned i0 = head[g];
        unsigned i1 = ((unsigned)(g + 1) < total) ? head[g + 1] : (unsigned)n;
        float acc[8] = {0, 0, 0, 0, 0, 0, 0, 0};
        for (unsigned t = i0 >> 4; t <= ((i1 - 1) >> 4); ++t) {
            int l = (int)((unsigned)g - tilefirst[t]);
            if (l >= 0 && l < 16) {
                const float* src = tilebuf + (size_t)t * 128 + (size_t)l * 8;
#pragma unroll
                for (int f = 0; f < 8; ++f) acc[f] += src[f];
            }
        }
        out_codes[g] = (long long)codes[i0];
        size_t go = (size_t)g * 8;
#pragma unroll
        for (int f = 0; f < 8; ++f) out_agg[go + f] = acc[f];
    } else {
        out_codes[g] = SENT_LL;
        size_t go = (size_t)g * 8;
#pragma unroll
        for (int f = 0; f < 8; ++f) out_agg[go + f] = 0.0f;
    }
}

__global__ void k_prep(const long long* __restrict__ ucodes, unsigned long long* __restrict__ codes_out,
                       int n) {
    int i = blockIdx.x * 256 + threadIdx.x;
    if (i >= n) return;
    long long u = ucodes[i];
    codes_out[i] = (u == SENT_LL) ? SENT_ULL : ((unsigned long long)u >> 3);
}

// ---------------- host side ----------------

extern "C" void kernel_launch(void* const* d_in, const int* in_sizes, int n_in,
                              void* d_out, int out_size, void* d_ws, size_t ws_size,
                              hipStream_t stream) {
    const float* cloud = (const float*)d_in[0];
    const float* leaf  = (const float*)d_in[1];
    const int N = in_sizes[0] / 4;
    const int numB = (N + 255) / 256;
    const int numTiles = (N + 15) / 16;
    const int MAX_DEPTH = 21;

    // workspace carve-out
    size_t off = 0;
    char* wsb = (char*)d_ws;
    auto take = [&](size_t bytes) -> char* {
        char* p = wsb + off;
        off += (bytes + 255) & ~(size_t)255;
        return p;
    };
    unsigned long long* codesA = (unsigned long long*)take((size_t)N * 8);
    unsigned long long* codesB = (unsigned long long*)take((size_t)N * 8);
    unsigned* idxA = (unsigned*)take((size_t)N * 4);
    unsigned* idxB = (unsigned*)take((size_t)N * 4);
    float*    feats = (float*)take((size_t)N * 32);
    unsigned* flagsB = (unsigned*)take((size_t)N * 4);
    unsigned* segB   = (unsigned*)take((size_t)N * 4);
    unsigned* headB  = (unsigned*)take((size_t)N * 4);
    unsigned* bsums  = (unsigned*)take((size_t)(numB + 1) * 4);
    float*    tilebuf = (float*)take((size_t)numTiles * 128 * 4);
    unsigned* tilefirst = (unsigned*)take((size_t)numTiles * 4);
    unsigned* rcounts = (unsigned*)take((size_t)256 * numB * 4);
    unsigned* rtmp    = (unsigned*)take((size_t)256 * numB * 4);
    unsigned* minb = (unsigned*)take(64);

    // output layout: mcs (21*N int64) | ags (21*N*8 f32) | cnts (21 int32)
    long long* out_mcs = (long long*)d_out;
    float* out_ags = (float*)((char*)d_out + (size_t)MAX_DEPTH * N * 8);
    int*   out_cnts = (int*)((char*)d_out + (size_t)MAX_DEPTH * N * 8 + (size_t)MAX_DEPTH * N * 8 * 4);

    // 1) origin = min(xyz) - 0.5*leaf
    k_init_min<<<1, 64, 0, stream>>>(minb);
    k_min<<<numB, 256, 0, stream>>>(cloud, N, minb);

    // 2) morton codes + feats
    k_morton<<<numB, 256, 0, stream>>>(cloud, leaf, minb, N, codesA, feats, idxA);

    // 3) radix sort: 8 passes of 8 bits, ping-pong A<->B, ends in A
    const int m = 256 * numB;
    const int numB2 = (m + 255) / 256;
    for (int pass = 0; pass < 8; ++pass) {
        unsigned long long* kin  = (pass & 1) ? codesB : codesA;
        unsigned long long* kout = (pass & 1) ? codesA : codesB;
        unsigned* iin  = (pass & 1) ? idxB : idxA;
        unsigned* iout = (pass & 1) ? idxA : idxB;
        int shift = pass * 8;
        k_hist<<<numB, 256, 0, stream>>>(kin, N, shift, rcounts, numB);
        // hierarchical exclusive scan of the 256*numB histogram
        k_bexscan<<<numB2, 256, 0, stream>>>(rcounts, rtmp, bsums, m);
        k_scan_single<<<1, 1024, 0, stream>>>(bsums, numB2);
        k_addscan<<<numB2, 256, 0, stream>>>(rtmp, bsums, rcounts, m);
        k_scatter<<<numB, 256, 0, stream>>>(kin, iin, N, shift, rcounts, numB, kout, iout);
    }

    // 4) 21 levels of unique-aggregate
    const unsigned long long* curCodes = codesA;
    const float* curVals = feats;
    const unsigned* curGather = idxA;   // level 0 gathers feats through sorted index
    for (int d = 0; d < MAX_DEPTH; ++d) {
        k_flags<<<numB, 256, 0, stream>>>(curCodes, flagsB, N);
        k_blockscan<<<numB, 256, 0, stream>>>(flagsB, segB, bsums, N);
        k_scan_single<<<1, 1024, 0, stream>>>(bsums, numB);
        k_addoff<<<numB, 256, 0, stream>>>(segB, bsums, flagsB, headB, N);
        k_tile_wmma<<<(numTiles + 7) / 8, 256, 0, stream>>>(curVals, curGather, segB, N,
                                                            tilebuf, tilefirst, numTiles);
        k_combine<<<numB, 256, 0, stream>>>(curCodes, segB, headB, tilebuf, tilefirst, N,
                                            out_mcs + (size_t)d * N,
                                            out_ags + (size_t)d * N * 8,
                                            out_cnts + d);
        if (d + 1 < MAX_DEPTH) {
            k_prep<<<numB, 256, 0, stream>>>(out_mcs + (size_t)d * N, codesB, N);
            curCodes = codesB;
            curVals = out_ags + (size_t)d * N * 8;
            curGather = nullptr;
        }
    }
}